// HeterogeneousGAT_38027640439232
// MI455X (gfx1250) — compile-verified
//
#include <hip/hip_runtime.h>
#include <math.h>

// ---------------------------------------------------------------------------
// Bipartite GATConv (user -> post), H=4 heads, C=64, concat -> 256 feats,
// then ReLU + [256,2] classifier + log_softmax.
//
// Pipeline:
//   1) gemm256_wmma   : xs = x_user @ W_src, xd = x_post @ W_dst  (fp32 WMMA)
//   2) att_scores     : a[n][h] = dot(Y[n, h*64:...], att[h])
//   3) init_ws        : m = -inf, denom = 0, agg = 0
//   4) edge_max       : segment max of leaky_relu(a_src[s]+a_dst[d]) (atomic)
//   5) edge_expsum    : ex = exp(e - m[d]); denom += ex (atomic)
//   6) edge_aggregate : agg[d] += (ex/denom[d]) * xs[s]  (atomic, 1 edge/block)
//   7) classify       : relu(agg+bias) @ W_cls + b_cls -> log_softmax
// ---------------------------------------------------------------------------

#define N_USER 200000
#define N_POST 100000
#define DIM    128
#define NEDGE  1000000
#define NH     4
#define HC     256   // H*C

typedef __attribute__((ext_vector_type(2))) float v2f;
typedef __attribute__((ext_vector_type(8))) float v8f;

// ---------------------------------------------------------------------------
// Y[nrows, 256] = X[nrows, 128] @ W[128, 256] using V_WMMA_F32_16X16X4_F32.
// One wave per 16x32 output slab (two 16x16 N-tiles -> two independent
// accumulator chains to cover WMMA RAW latency).
//
// A 16x4 f32 fragment (per ISA 7.12.2): lanes 0-15 hold K=0 (v0), K=1 (v1);
// lanes 16-31 hold K=2 (v0), K=3 (v1); M = lane%16.
// B 4x16 fragment mirrors A with N = lane%16.
// C/D: VGPR r holds M=r (lanes 0-15) / M=8+r (lanes 16-31), N = lane%16.
// ---------------------------------------------------------------------------
__global__ void gemm256_wmma(const float* __restrict__ X,
                             const float* __restrict__ W,
                             float* __restrict__ Y, int nrows) {
  const int lane = threadIdx.x & 31;
  const int wave = threadIdx.x >> 5;
  const int tile = blockIdx.x * (blockDim.x >> 5) + wave;
  const int mt = tile >> 3;            // row tile (16 rows)
  const int nt = tile & 7;             // column slab (32 cols)
  const int row0 = mt << 4;
  if (row0 >= nrows) return;           // wave-uniform: EXEC all-ones inside
  const int col0 = nt << 5;
  const int laneM = lane & 15;
  const int kOff = (lane >> 4) << 1;   // 0 for lanes 0-15, 2 for lanes 16-31

  const float* __restrict__ xrow = X + (size_t)(row0 + laneM) * DIM + kOff;
  const float* __restrict__ wcol = W + col0 + laneM;

  v8f acc0 = {};
  v8f acc1 = {};
  for (int kk = 0; kk < DIM; kk += 4) {
    v2f a;
    a.x = xrow[kk];
    a.y = xrow[kk + 1];
    const float* wp = wcol + (size_t)(kk + kOff) * HC;
    v2f b0; b0.x = wp[0];   b0.y = wp[HC];        // N-tile 0, K rows kOff/kOff+1
    v2f b1; b1.x = wp[16];  b1.y = wp[HC + 16];   // N-tile 1
    acc0 = __builtin_amdgcn_wmma_f32_16x16x4_f32(
        false, a, false, b0, (short)0, acc0, false, false);
    acc1 = __builtin_amdgcn_wmma_f32_16x16x4_f32(
        false, a, false, b1, (short)0, acc1, false, false);
  }

  float* yrow = Y + (size_t)row0 * HC + col0 + laneM;
#pragma unroll
  for (int r = 0; r < 8; ++r) {
    int mrow = (lane < 16) ? r : (r + 8);
    yrow[(size_t)mrow * HC] = acc0[r];
    yrow[(size_t)mrow * HC + 16] = acc1[r];
  }
}

// a[n*4+h] = sum_c Y[n][h*64+c] * att[h*64+c]
__global__ void att_scores(const float* __restrict__ Y,
                           const float* __restrict__ att,
                           float* __restrict__ a, int nrows) {
  int idx = blockIdx.x * blockDim.x + threadIdx.x;
  if (idx >= nrows * NH) return;
  int n = idx >> 2, h = idx & 3;
  const float* y = Y + (size_t)n * HC + h * 64;
  const float* at = att + h * 64;
  float s = 0.f;
#pragma unroll 8
  for (int c = 0; c < 64; ++c) s += y[c] * at[c];
  a[idx] = s;
}

__global__ void init_ws(float* __restrict__ m, float* __restrict__ denom,
                        float* __restrict__ agg) {
  int i = blockIdx.x * blockDim.x + threadIdx.x;
  const int n_mh = N_POST * NH;
  const int n_agg = N_POST * HC;
  if (i < n_mh) {
    m[i] = __int_as_float(0xff800000);  // -inf
    denom[i] = 0.f;
  }
  if (i < n_agg) agg[i] = 0.f;
}

__device__ __forceinline__ float leaky02(float v) {
  return v > 0.f ? v : 0.2f * v;
}

// Order-preserving float atomic max (init = -inf).
__device__ __forceinline__ void atomicMaxF(float* addr, float val) {
  if (!(val < 0.f)) {
    atomicMax((int*)addr, __float_as_int(val));
  } else {
    atomicMin((unsigned int*)addr, __float_as_uint(val));
  }
}

__global__ void edge_max(const int* __restrict__ src, const int* __restrict__ dst,
                         const float* __restrict__ a_src,
                         const float* __restrict__ a_dst,
                         float* __restrict__ m) {
  int e = blockIdx.x * blockDim.x + threadIdx.x;
  if (e >= NEDGE) return;
  int s = src[e], d = dst[e];
#pragma unroll
  for (int h = 0; h < NH; ++h) {
    float v = leaky02(a_src[s * NH + h] + a_dst[d * NH + h]);
    atomicMaxF(&m[d * NH + h], v);
  }
}

__global__ void edge_expsum(const int* __restrict__ src, const int* __restrict__ dst,
                            const float* __restrict__ a_src,
                            const float* __restrict__ a_dst,
                            const float* __restrict__ m,
                            float* __restrict__ denom, float* __restrict__ ex) {
  int e = blockIdx.x * blockDim.x + threadIdx.x;
  if (e >= NEDGE) return;
  int s = src[e], d = dst[e];
#pragma unroll
  for (int h = 0; h < NH; ++h) {
    float v = leaky02(a_src[s * NH + h] + a_dst[d * NH + h]);
    float t = __expf(v - m[d * NH + h]);
    ex[(size_t)e * NH + h] = t;
    atomicAdd(&denom[d * NH + h], t);
  }
}

// One block per edge, one thread per output channel (h = c/64).
__global__ void edge_aggregate(const int* __restrict__ src,
                               const int* __restrict__ dst,
                               const float* __restrict__ xs,
                               const float* __restrict__ ex,
                               const float* __restrict__ denom,
                               float* __restrict__ agg) {
  int e = blockIdx.x;
  int c = threadIdx.x;          // 0..255
  int h = c >> 6;
  int s = src[e], d = dst[e];
  float alpha = ex[(size_t)e * NH + h] / denom[d * NH + h];
  atomicAdd(&agg[(size_t)d * HC + c], alpha * xs[(size_t)s * HC + c]);
}

// One wave per post: relu(agg+bias) @ W_cls + b_cls -> log_softmax(2 classes).
__global__ void classify(const float* __restrict__ agg,
                         const float* __restrict__ bias,
                         const float* __restrict__ Wcls,
                         const float* __restrict__ bcls,
                         float* __restrict__ out) {
  int gw = (blockIdx.x * blockDim.x + threadIdx.x) >> 5;
  int lane = threadIdx.x & 31;
  if (gw >= N_POST) return;
  const float* row = agg + (size_t)gw * HC;
  float z0 = 0.f, z1 = 0.f;
#pragma unroll
  for (int c = lane; c < HC; c += 32) {
    float v = fmaxf(row[c] + bias[c], 0.f);
    z0 += v * Wcls[c * 2 + 0];
    z1 += v * Wcls[c * 2 + 1];
  }
#pragma unroll
  for (int off = 16; off > 0; off >>= 1) {
    z0 += __shfl_down(z0, off, 32);
    z1 += __shfl_down(z1, off, 32);
  }
  if (lane == 0) {
    z0 += bcls[0];
    z1 += bcls[1];
    float mx = fmaxf(z0, z1);
    float lse = mx + logf(__expf(z0 - mx) + __expf(z1 - mx));
    out[gw * 2 + 0] = z0 - lse;
    out[gw * 2 + 1] = z1 - lse;
  }
}

extern "C" void kernel_launch(void* const* d_in, const int* in_sizes, int n_in,
                              void* d_out, int out_size, void* d_ws, size_t ws_size,
                              hipStream_t stream) {
  const float* x_user  = (const float*)d_in[0];
  const float* x_post  = (const float*)d_in[1];
  const int*   src_idx = (const int*)d_in[2];
  const int*   dst_idx = (const int*)d_in[3];
  const float* W_src   = (const float*)d_in[4];
  const float* W_dst   = (const float*)d_in[5];
  const float* att_src = (const float*)d_in[6];
  const float* att_dst = (const float*)d_in[7];
  const float* bias    = (const float*)d_in[8];
  const float* W_cls   = (const float*)d_in[9];
  const float* b_cls   = (const float*)d_in[10];
  float* out = (float*)d_out;

  // Workspace layout (floats). Total ~108.4M floats (~434 MB).
  float* ws    = (float*)d_ws;
  float* xs    = ws;                               // N_USER*256
  float* xd    = xs    + (size_t)N_USER * HC;      // N_POST*256
  float* a_src = xd    + (size_t)N_POST * HC;      // N_USER*4
  float* a_dst = a_src + (size_t)N_USER * NH;      // N_POST*4
  float* m     = a_dst + (size_t)N_POST * NH;      // N_POST*4
  float* denom = m     + (size_t)N_POST * NH;      // N_POST*4
  float* ex    = denom + (size_t)N_POST * NH;      // E*4
  float* agg   = ex    + (size_t)NEDGE * NH;       // N_POST*256

  const int BLK = 256;               // 8 waves per block

  // 1) Projections via fp32 WMMA. tiles = (rows/16) * (256/32); 8 waves/block.
  {
    int tiles_u = (N_USER / 16) * (HC / 32);   // 100000
    int tiles_p = (N_POST / 16) * (HC / 32);   // 50000
    gemm256_wmma<<<tiles_u / 8, BLK, 0, stream>>>(x_user, W_src, xs, N_USER);
    gemm256_wmma<<<tiles_p / 8, BLK, 0, stream>>>(x_post, W_dst, xd, N_POST);
  }

  // 2) Per-head attention scores.
  att_scores<<<(N_USER * NH + BLK - 1) / BLK, BLK, 0, stream>>>(xs, att_src, a_src, N_USER);
  att_scores<<<(N_POST * NH + BLK - 1) / BLK, BLK, 0, stream>>>(xd, att_dst, a_dst, N_POST);

  // 3) Re-init accumulated buffers every call (graph replays don't re-poison).
  init_ws<<<(N_POST * HC + BLK - 1) / BLK, BLK, 0, stream>>>(m, denom, agg);

  // 4-6) Edge-parallel segment softmax + scatter aggregation.
  int eblocks = (NEDGE + BLK - 1) / BLK;
  edge_max<<<eblocks, BLK, 0, stream>>>(src_idx, dst_idx, a_src, a_dst, m);
  edge_expsum<<<eblocks, BLK, 0, stream>>>(src_idx, dst_idx, a_src, a_dst, m, denom, ex);
  edge_aggregate<<<NEDGE, HC, 0, stream>>>(src_idx, dst_idx, xs, ex, denom, agg);

  // 7) Classifier + log_softmax (one wave per post).
  classify<<<(N_POST * 32 + BLK - 1) / BLK, BLK, 0, stream>>>(agg, bias, W_cls, b_cls, out);
}